// RGCNModel_25331717112057
// MI455X (gfx1250) — compile-verified
//
#include <hip/hip_runtime.h>
#include <math.h>

// Problem sizes (match reference)
#define NN 100000LL   // nodes (divisible by 16)
#define RR 4LL        // relations
#define EE 500000LL   // edges per relation
#define DD 128        // feature dim (in = hid = out)

typedef __attribute__((ext_vector_type(2))) float v2f;
typedef __attribute__((ext_vector_type(8))) float v8f;

// ---------------------------------------------------------------- utilities
__global__ void zero_f32(float* __restrict__ p, long long n) {
  long long i = (long long)blockIdx.x * blockDim.x + threadIdx.x;
  long long stride = (long long)gridDim.x * blockDim.x;
  for (; i < n; i += stride) p[i] = 0.0f;
}

// degree counting: integer atomics into (reinterpreted) float buffers
__global__ void deg_count(const int* __restrict__ src, const int* __restrict__ dst,
                          int* __restrict__ degO, int* __restrict__ degI) {
  long long t = (long long)blockIdx.x * blockDim.x + threadIdx.x;
  if (t >= RR * EE) return;
  long long r = t / EE;
  atomicAdd(&degO[r * NN + src[t]], 1);
  atomicAdd(&degI[r * NN + dst[t]], 1);
}

// in-place: int count -> rsqrt(max(count,1))
__global__ void deg_to_rsqrt(float* __restrict__ p, long long n) {
  long long i = (long long)blockIdx.x * blockDim.x + threadIdx.x;
  if (i >= n) return;
  int c = ((const int*)p)[i];
  p[i] = rsqrtf(fmaxf((float)c, 1.0f));
}

// ------------------------------------------------- edge scatter (bandwidth bound)
// one wave32 per edge; each lane moves 4 contiguous floats (float4 load +
// 4x hardware global_atomic_add_f32)
__global__ void scatter_msgs(const float* __restrict__ x,      // [NN, DD]
                             const int* __restrict__ gather,   // node remap or null
                             const int* __restrict__ src,      // [RR*EE]
                             const int* __restrict__ dst,      // [RR*EE]
                             const float* __restrict__ rsO,    // [RR*NN]
                             float* __restrict__ agg) {        // [RR*NN*DD]
  long long gid = (long long)blockIdx.x * blockDim.x + threadIdx.x;
  long long t = gid >> 5;               // edge index (flat over relations)
  if (t >= RR * EE) return;
  int lane = threadIdx.x & 31;
  long long r = t / EE;
  int s = src[t];
  int d = dst[t];
  int sn = gather ? gather[s] : s;
  float sc = rsO[r * NN + s];
  float4 v = ((const float4*)(x + (long long)sn * DD))[lane];
  float* ap = agg + (r * NN + d) * DD + lane * 4;
  unsafeAtomicAdd(ap + 0, v.x * sc);
  unsafeAtomicAdd(ap + 1, v.y * sc);
  unsafeAtomicAdd(ap + 2, v.z * sc);
  unsafeAtomicAdd(ap + 3, v.w * sc);
}

// ------------------------------------------------- WMMA f32 GEMM (+bias, +tanh)
// out[n, j] = act( sum_r (agg[r][n,:]*rsI[r][n]) @ W[r][:, j] + sum_r bias[r][j] )
// block = 256 threads = 8 waves; 1 block = 16 node rows x 128 cols.
// Uses V_WMMA_F32_16X16X4_F32, accumulating over r and K in registers.
__global__ void __launch_bounds__(256) rgcn_gemm(
    const float* __restrict__ agg,   // [RR*NN*DD]
    const float* __restrict__ rsI,   // [RR*NN]
    const float* __restrict__ W,     // [RR, DD, DD]
    const float* __restrict__ bias,  // [RR, DD]
    float* __restrict__ out,         // [NN, DD]
    int do_tanh) {
  __shared__ float As[16][DD + 4];   // +4 pad: conflict-free column reads

  const int tile = blockIdx.x;            // node tile (16 rows)
  const int lane = threadIdx.x & 31;
  const int wave = threadIdx.x >> 5;      // 0..7 -> 16-col slice
  const int n16  = lane & 15;             // N index within tile / M for A reads
  const int half = lane >> 4;             // 0 or 1 (lane half)
  const int col  = (wave << 4) + n16;     // output column 0..127
  const int koff = half << 1;             // A/B fragment K offset: 0 or 2

  v8f acc = (v8f)0.0f;

  for (int r = 0; r < (int)RR; ++r) {
    // Cooperative stage of scaled A tile: 16 rows x 128 cols, 8 floats/thread.
    {
      const int row = threadIdx.x >> 4;          // 0..15
      const int cb  = (threadIdx.x & 15) << 3;   // 0..120 step 8
      const long long nrow = (long long)tile * 16 + row;
      const float sc = rsI[(long long)r * NN + nrow];
      const float* ap = agg + ((long long)r * NN + nrow) * DD + cb;
      float4 v0 = ((const float4*)ap)[0];
      float4 v1 = ((const float4*)ap)[1];
      float4 s0 = make_float4(v0.x * sc, v0.y * sc, v0.z * sc, v0.w * sc);
      float4 s1 = make_float4(v1.x * sc, v1.y * sc, v1.z * sc, v1.w * sc);
      *((float4*)&As[row][cb + 0]) = s0;
      *((float4*)&As[row][cb + 4]) = s1;
    }
    __syncthreads();

    const float* Wr = W + (long long)r * DD * DD;
    for (int k0 = 0; k0 < DD; k0 += 4) {
      const int kb = k0 + koff;
      // A fragment 16x4 f32: a[v] = A[m][2*(lane/16) + v]
      v2f a;
      a.x = As[n16][kb];
      a.y = As[n16][kb + 1];
      // B fragment 4x16 f32 (mirrored layout): b[v] = W[2*(lane/16)+v][col]
      v2f b;
      b.x = Wr[kb * DD + col];
      b.y = Wr[(kb + 1) * DD + col];
      acc = __builtin_amdgcn_wmma_f32_16x16x4_f32(
          /*neg_a=*/false, a, /*neg_b=*/false, b,
          /*c_mod=*/(short)0, acc, /*reuse_a=*/false, /*reuse_b=*/false);
    }
    __syncthreads();  // before next relation overwrites As
  }

  // bias summed over relations (depends only on column)
  float bj = 0.0f;
  for (int r = 0; r < (int)RR; ++r) bj += bias[r * DD + col];

  // C/D layout: VGPR v -> M = v + 8*(lane/16), N = lane%16
  const long long mbase = (long long)tile * 16 + (half << 3);
  for (int v = 0; v < 8; ++v) {
    float val = acc[v] + bj;
    if (do_tanh) val = tanhf(val);
    out[(mbase + v) * DD + col] = val;
  }
}

// ------------------------------------------------------------------ launcher
extern "C" void kernel_launch(void* const* d_in, const int* in_sizes, int n_in,
                              void* d_out, int out_size, void* d_ws, size_t ws_size,
                              hipStream_t stream) {
  const int*   input = (const int*)d_in[0];    // [NN]
  const int*   src   = (const int*)d_in[1];    // [RR, EE]
  const int*   dst   = (const int*)d_in[2];    // [RR, EE]
  const float* emb   = (const float*)d_in[3];  // [NN, DD]
  const float* W1    = (const float*)d_in[4];  // [RR, DD, DD]
  const float* b1    = (const float*)d_in[5];  // [RR, DD]
  const float* W2    = (const float*)d_in[6];  // [RR, DD, DD]
  const float* b2    = (const float*)d_in[7];  // [RR, DD]
  float* out = (float*)d_out;

  // workspace layout (floats)
  float* ws  = (float*)d_ws;
  float* rsO = ws;                             // RR*NN  (out-degree rsqrt)
  float* rsI = ws + RR * NN;                   // RR*NN  (in-degree rsqrt)
  float* agg = ws + 2 * RR * NN;               // RR*NN*DD
  float* h   = agg + RR * NN * DD;             // NN*DD

  const long long nDeg   = 2 * RR * NN;        // 800k
  const long long nAgg   = RR * NN * DD;       // 51.2M
  const long long nEdges = RR * EE;            // 2M

  // 1) degrees -> rsqrt tables (shared by both layers)
  zero_f32<<<(unsigned)((nDeg + 255) / 256), 256, 0, stream>>>(rsO, nDeg);
  deg_count<<<(unsigned)((nEdges + 255) / 256), 256, 0, stream>>>(
      src, dst, (int*)rsO, (int*)rsI);
  deg_to_rsqrt<<<(unsigned)((nDeg + 255) / 256), 256, 0, stream>>>(rsO, nDeg);

  // 2) layer 1: scatter emb (via input gather) -> agg, then GEMM + tanh -> h
  zero_f32<<<(unsigned)((nAgg + 255) / 256), 256, 0, stream>>>(agg, nAgg);
  scatter_msgs<<<(unsigned)((nEdges * 32 + 255) / 256), 256, 0, stream>>>(
      emb, input, src, dst, rsO, agg);
  rgcn_gemm<<<(unsigned)(NN / 16), 256, 0, stream>>>(agg, rsI, W1, b1, h, 1);

  // 3) layer 2: scatter h -> agg, then GEMM -> out
  zero_f32<<<(unsigned)((nAgg + 255) / 256), 256, 0, stream>>>(agg, nAgg);
  scatter_msgs<<<(unsigned)((nEdges * 32 + 255) / 256), 256, 0, stream>>>(
      h, nullptr, src, dst, rsO, agg);
  rgcn_gemm<<<(unsigned)(NN / 16), 256, 0, stream>>>(agg, rsI, W2, b2, out, 0);
}